// GIN_NodeWeightEncoder_11759620456599
// MI455X (gfx1250) — compile-verified
//
#include <hip/hip_runtime.h>

// CDNA5 / gfx1250: wave32, fp32 WMMA 16x16x4.
typedef __attribute__((ext_vector_type(2))) float v2f;
typedef __attribute__((ext_vector_type(8))) float v8f;

#define BN_EPS 1e-5f

// ---------------------------------------------------------------------------
// Edge scatter-add: agg[dst[e]] += X[src[e]]  (segment_sum), float4 vectorized.
// L2 (192MB) holds both X and agg entirely, so the atomics resolve in-cache.
// ---------------------------------------------------------------------------
template<int F>
__global__ void scatter_add_kernel(const float* __restrict__ X,
                                   const int* __restrict__ src,
                                   const int* __restrict__ dst,
                                   float* __restrict__ agg, int E) {
    constexpr int G = F / 4;  // float4 groups per node row
    long long t = (long long)blockIdx.x * blockDim.x + threadIdx.x;
    long long total = (long long)E * G;
    if (t >= total) return;
    int e = (int)(t / G);
    int g = (int)(t % G);
    int s = src[e], d = dst[e];
    const float4 v = *reinterpret_cast<const float4*>(&X[(long long)s * F + g * 4]);
    float* p = &agg[(long long)d * F + g * 4];
    atomicAdd(p + 0, v.x);
    atomicAdd(p + 1, v.y);
    atomicAdd(p + 2, v.z);
    atomicAdd(p + 3, v.w);
}

// ---------------------------------------------------------------------------
// fp32 WMMA GEMM: Y[M,NDIM] = relu( (A0 (+A1)) [M,K] @ W [K,NDIM] + bias )
//
// Block = 512 threads = 16 waves; one block owns one 16-row M-tile and all
// NDIM/16 == 16 column tiles, so the 16xK A-tile is loaded from global
// exactly ONCE per block (coalesced float4, with the x+agg add fused at
// staging time) into LDS, then all 16 waves feed their WMMA A-fragments with
// ds_load_b64. Row stride padded to K+4 dwords -> 16-lane fragment reads land
// on banks {0,4,...,60}: conflict-free.
//
// A frag (16x4, ISA 7.12.2): lanes 0-15 hold K=k,k+1; lanes 16-31 K=k+2,k+3.
// C/D frag: VGPR r -> row r (lanes 0-15) / row r+8 (lanes 16-31), col=lane&15.
// Requires M % 16 == 0 (50000 = 3125*16), NDIM == 16*blockDim.y, EXEC all-1s.
// ---------------------------------------------------------------------------
template<int K, int NDIM, bool ADD_A, bool STATS>
__global__ void wmma_gemm_bias_relu(const float* __restrict__ A0,
                                    const float* __restrict__ A1,
                                    const float* __restrict__ W,
                                    const float* __restrict__ bias,
                                    float* __restrict__ Y,
                                    float* __restrict__ colsum,
                                    float* __restrict__ colsumsq) {
    constexpr int LSTRIDE = K + 4;               // dword row stride (bank-safe)
    __shared__ float tileA[16 * LSTRIDE];

    const int lane  = threadIdx.x;               // 0..31
    const int wave  = threadIdx.y;               // 0..15 -> column tile
    const int tid   = wave * 32 + lane;          // 0..511
    const int tileM = blockIdx.x;

    // ---- stage A tile (16 x K) into LDS, coalesced, add fused ----
    const long long aBase = (long long)tileM * 16 * K;
    #pragma unroll
    for (int g = tid; g < 4 * K; g += 512) {     // 4*K float4 groups
        int arow = g / (K / 4);
        int acol = (g % (K / 4)) * 4;
        float4 v = *reinterpret_cast<const float4*>(&A0[aBase + (long long)arow * K + acol]);
        if (ADD_A) {
            float4 w = *reinterpret_cast<const float4*>(&A1[aBase + (long long)arow * K + acol]);
            v.x += w.x; v.y += w.y; v.z += w.z; v.w += w.w;
        }
        *reinterpret_cast<float4*>(&tileA[arow * LSTRIDE + acol]) = v;
    }
    __syncthreads();

    // ---- WMMA main loop ----
    const int r16  = lane & 15;
    const int hi   = lane >> 4;
    const int bCol = wave * 16 + r16;
    const float* aLds = &tileA[r16 * LSTRIDE + hi * 2];

    v8f acc = {0.f, 0.f, 0.f, 0.f, 0.f, 0.f, 0.f, 0.f};

    #pragma unroll 4
    for (int k = 0; k < K; k += 4) {
        float2 av = *reinterpret_cast<const float2*>(aLds + k);          // ds_load_b64
        v2f a; a.x = av.x; a.y = av.y;
        v2f b;
        b.x = W[(long long)(k + hi * 2    ) * NDIM + bCol];              // coalesced
        b.y = W[(long long)(k + hi * 2 + 1) * NDIM + bCol];
        acc = __builtin_amdgcn_wmma_f32_16x16x4_f32(
                  /*neg_a=*/false, a, /*neg_b=*/false, b,
                  /*c_mod=*/(short)0, acc, /*reuse_a=*/false, /*reuse_b=*/false);
    }

    // ---- epilogue: bias + ReLU (+ optional fused BN stats) ----
    const float bb = bias[bCol];
    float psum = 0.f, psq = 0.f;
    #pragma unroll
    for (int r = 0; r < 8; ++r) {
        long long row = (long long)tileM * 16 + r + hi * 8;
        float v = acc[r] + bb;
        v = v > 0.f ? v : 0.f;
        Y[row * NDIM + bCol] = v;
        psum += v;
        psq  += v * v;
    }
    if (STATS) {
        atomicAdd(&colsum[bCol], psum);
        atomicAdd(&colsumsq[bCol], psq);
    }
}

// ---------------------------------------------------------------------------
// BN fold: scale = gamma * rsqrt(var + eps); shift = beta - mean * scale
// ---------------------------------------------------------------------------
__global__ void bn_params_kernel(const float* __restrict__ colsum,
                                 const float* __restrict__ colsumsq,
                                 const float* __restrict__ gamma,
                                 const float* __restrict__ beta,
                                 float* __restrict__ scale,
                                 float* __restrict__ shift,
                                 int C, float invN) {
    int c = blockIdx.x * blockDim.x + threadIdx.x;
    if (c >= C) return;
    float mean = colsum[c] * invN;
    float var  = colsumsq[c] * invN - mean * mean;  // biased variance
    float s = gamma[c] * rsqrtf(var + BN_EPS);
    scale[c] = s;
    shift[c] = beta[c] - mean * s;
}

// BN apply, float4 vectorized (F multiple of 4)
template<int F>
__global__ void bn_apply_vec4(const float* __restrict__ Yin,
                              const float* __restrict__ scale,
                              const float* __restrict__ shift,
                              float* __restrict__ Yout, long long nvec) {
    long long t = (long long)blockIdx.x * blockDim.x + threadIdx.x;
    if (t >= nvec) return;
    int c = (int)((t * 4) % F);
    float4 v = *reinterpret_cast<const float4*>(&Yin[t * 4]);
    float4 s = *reinterpret_cast<const float4*>(&scale[c]);
    float4 h = *reinterpret_cast<const float4*>(&shift[c]);
    float4 o;
    o.x = v.x * s.x + h.x;
    o.y = v.y * s.y + h.y;
    o.z = v.z * s.z + h.z;
    o.w = v.w * s.w + h.w;
    *reinterpret_cast<float4*>(&Yout[t * 4]) = o;
}

// BN apply for the 2-wide head
__global__ void bn_apply2(const float* __restrict__ Yin,
                          const float* __restrict__ scale,
                          const float* __restrict__ shift,
                          float* __restrict__ Yout, int total) {
    int t = blockIdx.x * blockDim.x + threadIdx.x;
    if (t >= total) return;
    int c = t & 1;
    Yout[t] = Yin[t] * scale[c] + shift[c];
}

// ---------------------------------------------------------------------------
// 256 -> 2 projection + ReLU + BN-stat reduction (WMMA would waste 7/8 of the
// tile here; a VALU dot per (row, col) with LDS-atomic block reduction wins).
// ---------------------------------------------------------------------------
__global__ void linear_out2_relu_stats(const float* __restrict__ G1,
                                       const float* __restrict__ W4,
                                       const float* __restrict__ b4,
                                       float* __restrict__ outpre,
                                       float* __restrict__ colsum,
                                       float* __restrict__ colsumsq,
                                       int Nn) {
    __shared__ float s_sum[2], s_sq[2];
    int tid = threadIdx.x;
    if (tid < 2) { s_sum[tid] = 0.f; s_sq[tid] = 0.f; }
    __syncthreads();
    int idx = blockIdx.x * blockDim.x + tid;
    if (idx < Nn * 2) {
        int row = idx >> 1, c = idx & 1;
        float acc = b4[c];
        const float* g = &G1[(long long)row * 256];
        #pragma unroll 8
        for (int k = 0; k < 256; ++k)
            acc = fmaf(g[k], W4[k * 2 + c], acc);
        float v = acc > 0.f ? acc : 0.f;
        outpre[idx] = v;
        atomicAdd(&s_sum[c], v);        // ds_add_f32 (LDS atomic)
        atomicAdd(&s_sq[c], v * v);
    }
    __syncthreads();
    if (tid < 2) {
        atomicAdd(&colsum[tid], s_sum[tid]);
        atomicAdd(&colsumsq[tid], s_sq[tid]);
    }
}

// ---------------------------------------------------------------------------
extern "C" void kernel_launch(void* const* d_in, const int* in_sizes, int n_in,
                              void* d_out, int out_size, void* d_ws, size_t ws_size,
                              hipStream_t stream) {
    (void)n_in; (void)out_size; (void)ws_size;

    const float* x   = (const float*)d_in[0];
    const int*   ei  = (const int*)  d_in[1];
    const float* W1  = (const float*)d_in[2];
    const float* b1  = (const float*)d_in[3];
    const float* W2  = (const float*)d_in[4];
    const float* b2  = (const float*)d_in[5];
    const float* W3  = (const float*)d_in[6];
    const float* b3  = (const float*)d_in[7];
    const float* W4  = (const float*)d_in[8];
    const float* b4  = (const float*)d_in[9];
    const float* gm1 = (const float*)d_in[10];
    const float* bt1 = (const float*)d_in[11];
    const float* gm5 = (const float*)d_in[12];
    const float* bt5 = (const float*)d_in[13];

    const int FIN = 128, DIM = 256;
    const int Nn = in_sizes[0] / FIN;      // 50000 (multiple of 16)
    const int E  = in_sizes[1] / 2;        // 800000
    const int* src = ei;
    const int* dst = ei + E;

    // ---- workspace layout (floats) ----
    float* ws = (float*)d_ws;
    float* colsum1   = ws + 0;             // [256]
    float* colsumsq1 = ws + 256;           // [256]
    float* scale1    = ws + 512;           // [256]
    float* shift1    = ws + 768;           // [256]
    float* colsum5   = ws + 1024;          // [2]
    float* colsumsq5 = ws + 1028;          // [2]
    float* scale5    = ws + 1032;          // [2]
    float* shift5    = ws + 1036;          // [2]
    float* R0 = ws + 2048;                 // agg1       [N,128]
    float* R1 = R0 + (size_t)Nn * FIN;     // h1 -> agg2 [N,256]
    float* R2 = R1 + (size_t)Nn * DIM;     // h2 -> g1   [N,256]
    float* R3 = R2 + (size_t)Nn * DIM;     // outpre     [N,2]

    float* outp = (float*)d_out;           // [N,2]
    float* hid  = outp + (size_t)Nn * 2;   // hid_x_rep [N,256] lives in d_out

    // zero accumulators (graph-capturable stream memsets)
    hipMemsetAsync(ws, 0, 2048 * sizeof(float), stream);
    hipMemsetAsync(R0, 0, (size_t)Nn * FIN * sizeof(float), stream);

    const dim3 gemmBlk(32, 16);            // 16 waves: one per 16-col tile
    const dim3 gemmGrid(Nn / 16);          // one block per 16-row tile

    // 1) agg1 = segment_sum(x[src], dst)
    {
        long long total = (long long)E * (FIN / 4);
        scatter_add_kernel<128><<<(int)((total + 255) / 256), 256, 0, stream>>>(
            x, src, dst, R0, E);
    }
    // 2) h1 = relu((x + agg1) @ W1 + b1)          [WMMA f32, K=128]
    wmma_gemm_bias_relu<128, 256, true, false><<<gemmGrid, gemmBlk, 0, stream>>>(
        x, R0, W1, b1, R1, nullptr, nullptr);
    // 3) h2 = relu(h1 @ W2 + b2) + fused BN stats [WMMA f32, K=256]
    wmma_gemm_bias_relu<256, 256, false, true><<<gemmGrid, gemmBlk, 0, stream>>>(
        R1, nullptr, W2, b2, R2, colsum1, colsumsq1);
    // 4) BN1 fold + apply -> hid (written directly into d_out)
    bn_params_kernel<<<1, 256, 0, stream>>>(colsum1, colsumsq1, gm1, bt1,
                                            scale1, shift1, DIM, 1.0f / (float)Nn);
    {
        long long nvec = (long long)Nn * DIM / 4;
        bn_apply_vec4<256><<<(int)((nvec + 255) / 256), 256, 0, stream>>>(
            R2, scale1, shift1, hid, nvec);
    }
    // 5) agg2 = segment_sum(hid[src], dst)   (R1 reused; memset is stream-ordered)
    hipMemsetAsync(R1, 0, (size_t)Nn * DIM * sizeof(float), stream);
    {
        long long total = (long long)E * (DIM / 4);
        scatter_add_kernel<256><<<(int)((total + 255) / 256), 256, 0, stream>>>(
            hid, src, dst, R1, E);
    }
    // 6) g1 = relu((hid + agg2) @ W3 + b3)        [WMMA f32, K=256]
    wmma_gemm_bias_relu<256, 256, true, false><<<gemmGrid, gemmBlk, 0, stream>>>(
        hid, R1, W3, b3, R2, nullptr, nullptr);
    // 7) outpre = relu(g1 @ W4 + b4) + fused BN stats (2-wide head, VALU)
    linear_out2_relu_stats<<<(Nn * 2 + 255) / 256, 256, 0, stream>>>(
        R2, W4, b4, R3, colsum5, colsumsq5, Nn);
    // 8) BN5 fold + apply -> out
    bn_params_kernel<<<1, 32, 0, stream>>>(colsum5, colsumsq5, gm5, bt5,
                                           scale5, shift5, 2, 1.0f / (float)Nn);
    bn_apply2<<<(Nn * 2 + 255) / 256, 256, 0, stream>>>(R3, scale5, shift5,
                                                        outp, Nn * 2);
}